// QuantumSelfAttention_17171279250104
// MI455X (gfx1250) — compile-verified
//
#include <hip/hip_runtime.h>
#include <hip/hip_bf16.h>
#include <math.h>

typedef __attribute__((ext_vector_type(2))) float v2f;
typedef __attribute__((ext_vector_type(8))) float v8f;

#define BATCH 4
#define SEQ   128
#define DIM   256
#define NHEAD 8
#define DK    32

static __device__ __forceinline__ v8f wmma_f32_16x16x4(v2f a, v2f b, v8f c) {
  // D = A(16x4,f32) * B(4x16,f32) + C(16x16,f32); wave32, EXEC all-ones
  return __builtin_amdgcn_wmma_f32_16x16x4_f32(false, a, false, b, (short)0, c,
                                               false, false);
}

// One wave computes one 16x16 tile of O = X @ W^T + bias.
// X:(M,Kdim) row-major, W:(N,Kdim) row-major, O:(M,N) row-major.
// A layout (ISA 7.12.2, 32-bit A 16x4): lane%16 = M, VGPR pair holds
// contiguous K pair at K = k + 2*(lane/16). B mirrored with lane%16 = N.
static __device__ __forceinline__ void gemm_tile_body(
    const float* __restrict__ X, const float* __restrict__ W,
    const float* __restrict__ bias, float* __restrict__ O,
    int N, int Kdim) {
  const int lane = threadIdx.x;
  const int hi = lane >> 4;
  const int lo = lane & 15;
  const int m0 = blockIdx.x * 16;
  const int n0 = blockIdx.y * 16;
  const float* xrow = X + (size_t)(m0 + lo) * Kdim + 2 * hi;
  const float* wrow = W + (size_t)(n0 + lo) * Kdim + 2 * hi;
  v8f acc = {};
  for (int k = 0; k < Kdim; k += 4) {
    v2f a = *(const v2f*)(xrow + k);
    v2f b = *(const v2f*)(wrow + k);
    acc = wmma_f32_16x16x4(a, b, acc);
  }
  const float bvv = bias[n0 + lo];
#pragma unroll
  for (int r = 0; r < 8; ++r)
    O[(size_t)(m0 + r + 8 * hi) * N + (n0 + lo)] = acc[r] + bvv;
}

// grid (M/16, DIM/16, 3), block 32: fused Q/K/V projections.
__global__ void qkv_gemm(const float* __restrict__ x,
                         const float* __restrict__ Wq, const float* __restrict__ bq,
                         const float* __restrict__ Wk, const float* __restrict__ bk,
                         const float* __restrict__ Wv, const float* __restrict__ bv,
                         float* __restrict__ Q, float* __restrict__ K,
                         float* __restrict__ V) {
  const float* W; const float* bb; float* O;
  if (blockIdx.z == 0)      { W = Wq; bb = bq; O = Q; }
  else if (blockIdx.z == 1) { W = Wk; bb = bk; O = K; }
  else                      { W = Wv; bb = bv; O = V; }
  gemm_tile_body(x, W, bb, O, DIM, DIM);
}

// Output projection: out = ctx @ Wo^T + bo
__global__ void out_gemm(const float* __restrict__ ctx,
                         const float* __restrict__ Wo,
                         const float* __restrict__ bo,
                         float* __restrict__ out) {
  gemm_tile_body(ctx, Wo, bo, out, DIM, DIM);
}

// Per (b,h,s): normalize first 4 head-dims of Q (or K) to [0,pi], store
// cos(ang/2), sin(ang/2) per qubit. feat row index = (b*H+h)*S + s.
__global__ void angle_features(const float* __restrict__ Q,
                               const float* __restrict__ K,
                               float* __restrict__ qfeat,
                               float* __restrict__ kfeat) {
  const int total = BATCH * NHEAD * SEQ;
  int tid = blockIdx.x * blockDim.x + threadIdx.x;
  if (tid >= 2 * total) return;
  const float* src = (tid < total) ? Q : K;
  float* dst       = (tid < total) ? qfeat : kfeat;
  int idx = (tid < total) ? tid : (tid - total);
  int s = idx % SEQ;
  int h = (idx / SEQ) % NHEAD;
  int b = idx / (SEQ * NHEAD);
  const float* p = src + (size_t)(b * SEQ + s) * DIM + h * DK;
  float v0 = p[0], v1 = p[1], v2 = p[2], v3 = p[3];
  float mn = fminf(fminf(v0, v1), fminf(v2, v3));
  float mx = fmaxf(fmaxf(v0, v1), fmaxf(v2, v3));
  float scale = 3.14159265358979323846f / (mx - mn + 1e-8f);
  float vv[4] = {v0, v1, v2, v3};
  float* o = dst + (size_t)idx * 8;
#pragma unroll
  for (int w = 0; w < 4; ++w) {
    float half_ang = (vv[w] - mn) * scale * 0.5f;
    o[2 * w]     = cosf(half_ang);
    o[2 * w + 1] = sinf(half_ang);
  }
}

// One block (128 threads) per (b,h,i) row: each thread evaluates the fixed
// 4-qubit circuit for pair (i, j=threadIdx.x), then fused softmax over j.
// State index n = q0*8 + q1*4 + q2*2 + q3.
__global__ void scores_softmax(const float* __restrict__ qfeat,
                               const float* __restrict__ kfeat,
                               const float* __restrict__ params,
                               float* __restrict__ attn) {
  const int j = threadIdx.x;       // 0..127
  const int row = blockIdx.x;      // (b*H+h)*S + i
  const int bh = row / SEQ;
  __shared__ float red[SEQ];

  float qf[8], kf[8];
  const float* qp = qfeat + (size_t)row * 8;
  const float* kp = kfeat + (size_t)(bh * SEQ + j) * 8;
#pragma unroll
  for (int t = 0; t < 8; ++t) { qf[t] = qp[t]; kf[t] = kp[t]; }

  float ct[4], st[4];
#pragma unroll
  for (int w = 0; w < 4; ++w) {
    float th = params[w] * 0.5f;
    ct[w] = cosf(th);
    st[w] = sinf(th);
  }

  // Per-qubit kets: amp0 = cos(q/2) e^{-ik/2}; amp1 = -i sin(q/2) e^{+ik/2}
  float wre[4][2], wim[4][2];
#pragma unroll
  for (int w = 0; w < 4; ++w) {
    float cq = qf[2 * w], sq = qf[2 * w + 1];
    float ck = kf[2 * w], sk = kf[2 * w + 1];
    wre[w][0] = cq * ck;  wim[w][0] = -cq * sk;
    wre[w][1] = sq * sk;  wim[w][1] = -sq * ck;
  }

  // Rank-1 product state (16 complex amplitudes, fully unrolled)
  float sre[16], sim[16];
#pragma unroll
  for (int n = 0; n < 16; ++n) {
    int q0 = (n >> 3) & 1, q1 = (n >> 2) & 1, q2 = (n >> 1) & 1, q3 = n & 1;
    float r1 = wre[0][q0] * wre[1][q1] - wim[0][q0] * wim[1][q1];
    float i1 = wre[0][q0] * wim[1][q1] + wim[0][q0] * wre[1][q1];
    float r2 = r1 * wre[2][q2] - i1 * wim[2][q2];
    float i2 = r1 * wim[2][q2] + i1 * wre[2][q2];
    sre[n] = r2 * wre[3][q3] - i2 * wim[3][q3];
    sim[n] = r2 * wim[3][q3] + i2 * wre[3][q3];
  }

  // RX(theta): new0 = c*s0 - i*s*s1 ; new1 = -i*s*s0 + c*s1 (real shuffles)
  auto rx = [&](int mask, float c, float sg) {
#pragma unroll
    for (int n = 0; n < 16; ++n) {
      if (n & mask) continue;
      int n1 = n | mask;
      float are = sre[n], aim = sim[n], bre = sre[n1], bim = sim[n1];
      sre[n]  = c * are + sg * bim;
      sim[n]  = c * aim - sg * bre;
      sre[n1] = c * bre + sg * aim;
      sim[n1] = c * bim - sg * are;
    }
  };
  // CNOT = register permutation (swap target bit where control bit set)
  auto cnot = [&](int cmask, int tmask) {
#pragma unroll
    for (int n = 0; n < 16; ++n) {
      if (!(n & cmask) || (n & tmask)) continue;
      int n1 = n | tmask;
      float tr = sre[n]; sre[n] = sre[n1]; sre[n1] = tr;
      float ti = sim[n]; sim[n] = sim[n1]; sim[n1] = ti;
    }
  };

  rx(8, ct[0], st[0]); cnot(8, 4);
  rx(4, ct[1], st[1]); cnot(4, 2);
  rx(2, ct[2], st[2]); cnot(2, 1);
  rx(1, ct[3], st[3]); cnot(1, 8);

  // <Z> on qubit 0
  float score = 0.f;
#pragma unroll
  for (int n = 0; n < 16; ++n) {
    float p = sre[n] * sre[n] + sim[n] * sim[n];
    score += (n < 8) ? p : -p;
  }

  // fused softmax over j (scale = 1/sqrt(dk), dk = 32)
  float val = score * 0.17677669529663688f;
  red[j] = val;
  __syncthreads();
  for (int off = 64; off > 0; off >>= 1) {
    if (j < off) red[j] = fmaxf(red[j], red[j + off]);
    __syncthreads();
  }
  float mval = red[0];
  __syncthreads();
  float e = __expf(val - mval);
  red[j] = e;
  __syncthreads();
  for (int off = 64; off > 0; off >>= 1) {
    if (j < off) red[j] += red[j + off];
    __syncthreads();
  }
  attn[(size_t)row * SEQ + j] = e / red[0];
}

// ctx[b,i,h*32+d] = sum_j attn[b,h,i,j] * V[b,j,h*32+d]
// grid (SEQ/16, DK/16, B*H), block 32 (one wave per 16x16 tile).
__global__ void av_gemm(const float* __restrict__ attn,
                        const float* __restrict__ V,
                        float* __restrict__ ctx) {
  const int lane = threadIdx.x;
  const int hi = lane >> 4;
  const int lo = lane & 15;
  const int m0 = blockIdx.x * 16;
  const int n0 = blockIdx.y * 16;
  const int bh = blockIdx.z;
  const int b = bh / NHEAD, h = bh % NHEAD;
  const float* arow = attn + ((size_t)bh * SEQ + m0 + lo) * SEQ + 2 * hi;
  const float* vbase = V + (size_t)b * SEQ * DIM + h * DK + n0 + lo;
  v8f acc = {};
  for (int k = 0; k < SEQ; k += 4) {
    v2f a = *(const v2f*)(arow + k);
    int ka = k + 2 * hi;
    v2f bb;
    bb.x = vbase[(size_t)ka * DIM];
    bb.y = vbase[(size_t)(ka + 1) * DIM];
    acc = wmma_f32_16x16x4(a, bb, acc);
  }
#pragma unroll
  for (int r = 0; r < 8; ++r) {
    int i = m0 + r + 8 * hi;
    ctx[(size_t)(b * SEQ + i) * DIM + h * DK + (n0 + lo)] = acc[r];
  }
}

extern "C" void kernel_launch(void* const* d_in, const int* in_sizes, int n_in,
                              void* d_out, int out_size, void* d_ws, size_t ws_size,
                              hipStream_t stream) {
  const float* x      = (const float*)d_in[0];
  const float* Wq     = (const float*)d_in[1];
  const float* bq     = (const float*)d_in[2];
  const float* Wk     = (const float*)d_in[3];
  const float* bk     = (const float*)d_in[4];
  const float* Wv     = (const float*)d_in[5];
  const float* bv     = (const float*)d_in[6];
  const float* Wo     = (const float*)d_in[7];
  const float* bo     = (const float*)d_in[8];
  const float* params = (const float*)d_in[9];

  float* ws = (float*)d_ws;
  const size_t nQ = (size_t)BATCH * SEQ * DIM;          // 131072
  const size_t nF = (size_t)BATCH * NHEAD * SEQ * 8;    // 32768
  float* Q     = ws;
  float* K     = Q + nQ;
  float* V     = K + nQ;
  float* qfeat = V + nQ;
  float* kfeat = qfeat + nF;
  float* attn  = kfeat + nF;
  float* ctx   = attn + (size_t)BATCH * NHEAD * SEQ * SEQ;
  float* out   = (float*)d_out;

  // 1) fused QKV projections (WMMA f32)
  dim3 g1(BATCH * SEQ / 16, DIM / 16, 3);
  qkv_gemm<<<g1, 32, 0, stream>>>(x, Wq, bq, Wk, bk, Wv, bv, Q, K, V);

  // 2) angle normalization -> cos/sin features
  int totF = 2 * BATCH * NHEAD * SEQ;
  angle_features<<<(totF + 255) / 256, 256, 0, stream>>>(Q, K, qfeat, kfeat);

  // 3) quantum scores + fused softmax
  scores_softmax<<<BATCH * NHEAD * SEQ, SEQ, 0, stream>>>(qfeat, kfeat, params, attn);

  // 4) attn @ V (WMMA f32), written directly in (B,S,D) layout
  dim3 g2(SEQ / 16, DK / 16, BATCH * NHEAD);
  av_gemm<<<g2, 32, 0, stream>>>(attn, V, ctx);

  // 5) output projection (WMMA f32)
  dim3 g3(BATCH * SEQ / 16, DIM / 16, 1);
  out_gemm<<<g3, 32, 0, stream>>>(ctx, Wo, bo, out);
}